// KNN_5523327943250
// MI455X (gfx1250) — compile-verified
//
#include <hip/hip_runtime.h>
#include <math.h>

typedef __attribute__((ext_vector_type(2))) float v2f;
typedef __attribute__((ext_vector_type(8))) float v8f;

#define NB    4
#define NQ    2048
#define NR    16384
#define DIM   64
#define KNN   16
#define QT    16          // queries per block
#define RT    16          // refs per tile
#define NWAVE 4
#define TPB   (NWAVE * 32)
#define ROWP  68          // padded row stride in floats (64 + 4) -> conflict-free b64 reads

__global__ __launch_bounds__(TPB) void knn_5523_kernel(
    const float* __restrict__ ref,
    const float* __restrict__ query,
    float* __restrict__ outD,
    long long* __restrict__ outI)
{
    __shared__ float sQ[QT * ROWP];            //  4.3 KB
    __shared__ float sQ2[QT];
    __shared__ float sRef[NWAVE][RT * ROWP];   // 17.4 KB
    __shared__ float sR2[NWAVE][RT];
    __shared__ float sD2[NWAVE][QT * 17];      //  4.3 KB (17-float rows: conflict-free)
    __shared__ float sCandD[QT * 128];         //  8 KB
    __shared__ int   sCandI[QT * 128];         //  8 KB

    const int tid  = threadIdx.x;
    const int lane = tid & 31;
    const int wave = tid >> 5;
    const int half = lane >> 4;   // 0: lanes 0-15, 1: lanes 16-31
    const int row  = lane & 15;

    const int batch = blockIdx.x / (NQ / QT);
    const int qt    = blockIdx.x % (NQ / QT);

    const float* Qbase = query + ((size_t)batch * NQ + (size_t)qt * QT) * DIM;
    const float* Rbase = ref + (size_t)batch * NR * DIM;

    // ---- stage Q tile (16x64) into LDS, padded rows; coalesced float4 ----
    for (int c = tid; c < QT * (DIM / 4); c += TPB) {
        int r = c >> 4, c4 = c & 15;           // c == r*16 + c4 (row-major contiguous)
        float4 v = ((const float4*)Qbase)[c];
        *(float4*)&sQ[r * ROWP + c4 * 4] = v;
    }
    __syncthreads();
    if (tid < QT) {
        float s = 0.f;
        #pragma unroll
        for (int k4 = 0; k4 < DIM / 4; ++k4) {
            float4 v = *(const float4*)&sQ[tid * ROWP + k4 * 4];
            s += v.x * v.x + v.y * v.y + v.z * v.z + v.w * v.w;
        }
        sQ2[tid] = s;
    }
    __syncthreads();

    // ---- A fragments for all 16 K-steps, kept live for the whole block ----
    // A 16x4 f32 layout: lane L -> M = L%16; VGPRs hold K = 4s + 2*half, +1
    v2f afrag[16];
    #pragma unroll
    for (int s = 0; s < 16; ++s)
        afrag[s] = *(const v2f*)&sQ[row * ROWP + 4 * s + 2 * half];

    // ---- per-lane top-K: this lane covers query `row`, its wave/half's refs ----
    float bd[KNN]; int bi[KNN];
    #pragma unroll
    for (int j = 0; j < KNN; ++j) { bd[j] = 3.4e38f; bi[j] = 0; }

    const int tilesPerWave = (NR / RT) / NWAVE;   // 256
    for (int t = 0; t < tilesPerWave; ++t) {
        const int tile = t * NWAVE + wave;
        const float* Rt = Rbase + (size_t)tile * RT * DIM;

        // coalesced float4 global -> LDS (this wave's buffer), padded rows
        #pragma unroll
        for (int j = 0; j < (RT * DIM / 4) / 32; ++j) {   // 8 iters
            int c = lane + 32 * j;
            int r = c >> 4, c4 = c & 15;
            float4 v = ((const float4*)Rt)[c];
            *(float4*)&sRef[wave][r * ROWP + c4 * 4] = v;
        }
        if (t + 1 < tilesPerWave) {
            // prefetch next tile: 32 lanes x 128B == 4KB tile
            __builtin_prefetch(Rbase + (size_t)(tile + NWAVE) * RT * DIM + lane * 32, 0, 0);
        }
        __syncthreads();

        // r2 for this tile (lanes 0..15 of each wave)
        if (half == 0) {
            float s = 0.f;
            #pragma unroll
            for (int k4 = 0; k4 < DIM / 4; ++k4) {
                float4 v = *(const float4*)&sRef[wave][row * ROWP + k4 * 4];
                s += v.x * v.x + v.y * v.y + v.z * v.z + v.w * v.w;
            }
            sR2[wave][row] = s;
        }

        // ---- WMMA: acc(16x16) = Qtile(16x64) x Rtile^T(64x16), 16 chained k=4 steps
        v8f acc = {};
        #pragma unroll
        for (int s = 0; s < 16; ++s) {
            v2f b = *(const v2f*)&sRef[wave][row * ROWP + 4 * s + 2 * half];
            acc = __builtin_amdgcn_wmma_f32_16x16x4_f32(
                      false, afrag[s], false, b, (short)0, acc, false, false);
        }
        __syncthreads();

        // d2 tile -> LDS. C/D layout: lane holds N = row, VGPR i -> M = i + 8*half
        #pragma unroll
        for (int i = 0; i < 8; ++i) {
            int m = i + 8 * half;
            float d2 = sQ2[m] + sR2[wave][row] - 2.0f * acc[i];
            sD2[wave][m * 17 + row] = fmaxf(d2, 0.0f);
        }
        __syncthreads();

        // selection: lane owns query `row`; half selects refs n in [8*half, 8*half+8)
        #pragma unroll
        for (int j = 0; j < 8; ++j) {
            int n = 8 * half + j;
            float d = sD2[wave][row * 17 + n];
            if (d < bd[KNN - 1]) {
                bd[KNN - 1] = d; bi[KNN - 1] = tile * RT + n;
                #pragma unroll
                for (int u = KNN - 1; u > 0; --u) {
                    if (bd[u] < bd[u - 1]) {
                        float td = bd[u]; bd[u] = bd[u - 1]; bd[u - 1] = td;
                        int   ti = bi[u]; bi[u] = bi[u - 1]; bi[u - 1] = ti;
                    }
                }
            }
        }
        __syncthreads();
    }

    // ---- dump 8 sorted 16-lists per query to LDS ----
    {
        int base = row * 128 + (wave * 2 + half) * KNN;
        #pragma unroll
        for (int j = 0; j < KNN; ++j) { sCandD[base + j] = bd[j]; sCandI[base + j] = bi[j]; }
    }
    __syncthreads();

    // ---- final merge: 16 threads, one query each, scan 128 candidates ----
    if (tid < QT) {
        float fd[KNN]; int fi[KNN];
        #pragma unroll
        for (int j = 0; j < KNN; ++j) { fd[j] = 3.4e38f; fi[j] = 0; }
        for (int c = 0; c < 128; ++c) {
            float d  = sCandD[tid * 128 + c];
            int   id = sCandI[tid * 128 + c];
            if (d < fd[KNN - 1]) {
                fd[KNN - 1] = d; fi[KNN - 1] = id;
                #pragma unroll
                for (int u = KNN - 1; u > 0; --u) {
                    if (fd[u] < fd[u - 1]) {
                        float td = fd[u]; fd[u] = fd[u - 1]; fd[u - 1] = td;
                        int   ti = fi[u]; fi[u] = fi[u - 1]; fi[u - 1] = ti;
                    }
                }
            }
        }
        size_t obase = ((size_t)batch * NQ + (size_t)qt * QT + tid) * KNN;
        #pragma unroll
        for (int j = 0; j < KNN; ++j) {
            outD[obase + j] = sqrtf(fd[j]);
            outI[obase + j] = (long long)fi[j];
        }
    }
}

extern "C" void kernel_launch(void* const* d_in, const int* in_sizes, int n_in,
                              void* d_out, int out_size, void* d_ws, size_t ws_size,
                              hipStream_t stream) {
    const float* ref   = (const float*)d_in[0];   // [4, 16384, 64] f32
    const float* query = (const float*)d_in[1];   // [4,  2048, 64] f32

    float* outD = (float*)d_out;                                  // [4,2048,16] f32
    long long* outI = (long long*)(outD + (size_t)NB * NQ * KNN); // [4,2048,16] i64

    dim3 grid(NB * (NQ / QT));   // 512 blocks, one 16-query tile each
    knn_5523_kernel<<<grid, TPB, 0, stream>>>(ref, query, outD, outI);
}